// ReactionCenterMPNN_62732292326001
// MI455X (gfx1250) — compile-verified
//
#include <hip/hip_runtime.h>
#include <math.h>

// Problem constants (from the reference)
#define N_NODES 50000
#define N_EDGES 100000
#define FN      22      // atom features
#define HD      64      // hidden dim
#define BN_EPS  1e-5f

typedef __attribute__((ext_vector_type(2))) float v2f;
typedef __attribute__((ext_vector_type(4))) float v4f;
typedef __attribute__((ext_vector_type(8))) float v8f;

__device__ __forceinline__ void atomAddF32(float* p, float v) {
  // hardware global_atomic_add_f32 (no CAS loop)
  unsafeAtomicAdd(p, v);
}

// ---------------- zero the stats scratch ----------------
__global__ void k_zero(float* __restrict__ p, int n) {
  int i = blockIdx.x * blockDim.x + threadIdx.x;
  if (i < n) p[i] = 0.0f;
}

// ---------------- pack We/be into lane-major WMMA tile layout ----------------
// packed[(t*32 + l)*4 + j]:
//   j=0,1 : We[(l>>4)*2 + j][t*16 + (l&15)]   (B-matrix pair for lane l, tile t)
//   j=2   : be[t*16 + (l&15)]                 (column bias)
//   j=3   : 0
// One v4f load per lane per tile in the edge kernel (fully coalesced 512B/wave).
__global__ void k_pack_we(const float* __restrict__ We,   // [4, NCOL]
                          const float* __restrict__ be,   // [NCOL]
                          float* __restrict__ packed, int ncol) {
  const int idx = blockIdx.x * blockDim.x + threadIdx.x;
  if (idx >= 8 * ncol) return;               // 4 floats * 32 lanes * (ncol/16) tiles
  const int j = idx & 3;
  const int l = (idx >> 2) & 31;
  const int t = idx >> 7;
  const int col = t * 16 + (l & 15);
  float v;
  if (j < 2)      v = We[((l >> 4) * 2 + j) * ncol + col];
  else if (j == 2) v = be[col];
  else             v = 0.0f;
  packed[idx] = v;
}

// ---------------- out[r,0:64] = X[r,0:K] @ W[K,64] + bias ----------------
// One wave per 16 rows; 4 resident C tiles cover all 64 output columns.
// v_wmma_f32_16x16x4_f32, K stepped by 4 with tail masking (K=22 case).
template<int K>
__global__ void k_root_gemm(const float* __restrict__ X,
                            const float* __restrict__ W,
                            const float* __restrict__ bias,
                            float* __restrict__ out, int nrows) {
  const int lane = threadIdx.x & 31;
  const int wave = threadIdx.x >> 5;
  const int rb = (blockIdx.x * 8 + wave) * 16;
  if (rb >= nrows) return;

  const int n  = lane & 15;         // output column within tile / A row sel
  const int kh = (lane >> 4) * 2;   // K pair owned by this half-wave

  v8f c[4];
#pragma unroll
  for (int ot = 0; ot < 4; ++ot) {
    float bv = bias[ot * 16 + n];
#pragma unroll
    for (int r = 0; r < 8; ++r) c[ot][r] = bv;
  }

  const int m = rb + n;             // A-matrix row for this lane
  for (int kk = 0; kk < K; kk += 4) {
    const int k0 = kk + kh;
    v2f a;
    a.x = (k0     < K) ? X[m * K + k0]     : 0.0f;
    a.y = (k0 + 1 < K) ? X[m * K + k0 + 1] : 0.0f;
#pragma unroll
    for (int ot = 0; ot < 4; ++ot) {
      v2f b;
      b.x = (k0     < K) ? W[(k0    ) * HD + ot * 16 + n] : 0.0f;
      b.y = (k0 + 1 < K) ? W[(k0 + 1) * HD + ot * 16 + n] : 0.0f;
      c[ot] = __builtin_amdgcn_wmma_f32_16x16x4_f32(
          false, a, false, b, (short)0, c[ot], false, false);
    }
  }

  // C/D layout: vgpr r, lanes<16 -> row rb+r, lanes>=16 -> row rb+8+r
  const int eh = 8 * (lane >> 4);
#pragma unroll
  for (int ot = 0; ot < 4; ++ot)
#pragma unroll
    for (int r = 0; r < 8; ++r)
      out[(rb + eh + r) * HD + ot * 16 + n] = c[ot][r];
}

// ---------------- fused NNConv edge pipeline ----------------
// One wave per 16 edges. For each input channel i and 16-col slab t = i*4+ot:
//   Wtile = relu( A(edge_attr 16x4) x B(packed We 4x16) + be )  via WMMA f32 16x16x4
//   msg_tile += x[src[e], i] * Wtile     (scalar is half-wave uniform, LDS broadcast)
// Then 64 atomic f32 adds per edge into out[dst].
template<int IN>
__global__ void k_edge_msg(const float* __restrict__ xfeat,      // [N, IN]
                           const float* __restrict__ ea,         // [E, 4]
                           const int*   __restrict__ eidx,       // [2, E]
                           const float* __restrict__ packed,     // packed We+be
                           float* __restrict__ out, int nE) {
  const int lane = threadIdx.x & 31;
  const int wave = threadIdx.x >> 5;
  const int eb = (blockIdx.x * 8 + wave) * 16;
  const bool active = eb < nE;   // wave-uniform; EXEC stays all-ones for WMMA

  // s_x layout: [i][edge] so the 8 per-channel scalars are contiguous (b128 loads)
  __shared__ __align__(16) float s_x[8][IN * 16];
  __shared__ int s_src[8][16];
  __shared__ int s_dst[8][16];

  if (active && lane < 16) {
    s_src[wave][lane] = eidx[eb + lane];
    s_dst[wave][lane] = eidx[nE + eb + lane];
  }
  __syncthreads();
  if (active) {
    for (int idx = lane; idx < 16 * IN; idx += 32) {
      const int e = idx & 15;
      const int i = idx >> 4;
      s_x[wave][idx] = xfeat[s_src[wave][e] * IN + i];   // gather x[src]
    }
  }
  __syncthreads();
  if (!active) return;

  const int n  = lane & 15;
  const int kh = (lane >> 4) * 2;
  const int eh = 8 * (lane >> 4);

  // A matrix: 16 edges x 4 edge-features (K=4 == F_EDGE: one WMMA per tile)
  const v2f a = *(const v2f*)(ea + (eb + n) * 4 + kh);   // contiguous pair -> b64

  const v4f* pk = (const v4f*)packed;
  const v4f* xp = (const v4f*)(&s_x[wave][0]);
  const int  xo = eh >> 2;                                // 0 or 2

  float macc[4][8];
#pragma unroll
  for (int ot = 0; ot < 4; ++ot)
#pragma unroll
    for (int r = 0; r < 8; ++r) macc[ot][r] = 0.0f;

  for (int i = 0; i < IN; ++i) {
    const v4f x0 = xp[i * 4 + xo];          // x[src[e], i] for rows 0..3 (half-wave)
    const v4f x1 = xp[i * 4 + xo + 1];      // rows 4..7
    float xs[8];
#pragma unroll
    for (int r = 0; r < 4; ++r) { xs[r] = x0[r]; xs[r + 4] = x1[r]; }
#pragma unroll
    for (int ot = 0; ot < 4; ++ot) {
      const int t = i * 4 + ot;             // global 16-col tile index
      const v4f p = pk[t * 32 + lane];      // {We[kh][c], We[kh+1][c], be[c], 0}
      v2f b; b.x = p[0]; b.y = p[1];
      v8f c;
#pragma unroll
      for (int r = 0; r < 8; ++r) c[r] = p[2];            // bias splat (dual-issue movs)
      c = __builtin_amdgcn_wmma_f32_16x16x4_f32(
          false, a, false, b, (short)0, c, false, false);
#pragma unroll
      for (int r = 0; r < 8; ++r)
        macc[ot][r] = fmaf(xs[r], fmaxf(c[r], 0.0f), macc[ot][r]);
    }
  }

  // scatter-add messages into out[dst]
#pragma unroll
  for (int ot = 0; ot < 4; ++ot)
#pragma unroll
    for (int r = 0; r < 8; ++r) {
      const int d = s_dst[wave][eh + r];
      atomAddF32(&out[d * HD + ot * 16 + n], macc[ot][r]);
    }
}

// ---------------- BatchNorm statistics (per-channel sum / sumsq) ----------------
__global__ void k_bn_stats(const float* __restrict__ h,
                           float* __restrict__ sum, float* __restrict__ sq,
                           int nrows) {
  const int c  = threadIdx.x & 63;
  const int rw = threadIdx.x >> 6;       // 0..3 rows per block per iter
  float s = 0.0f, q = 0.0f;
  for (int r = blockIdx.x * 4 + rw; r < nrows; r += gridDim.x * 4) {
    const float v = h[r * HD + c];
    s += v; q += v * v;
  }
  __shared__ float ls[256], lq[256];
  ls[threadIdx.x] = s; lq[threadIdx.x] = q;
  __syncthreads();
  if (threadIdx.x < 64) {
    s = ls[c] + ls[c + 64] + ls[c + 128] + ls[c + 192];
    q = lq[c] + lq[c + 64] + lq[c + 128] + lq[c + 192];
    atomAddF32(&sum[c], s);
    atomAddF32(&sq[c],  q);
  }
}

// ---------------- BatchNorm apply + ReLU ----------------
__global__ void k_bn_apply(const float* __restrict__ in,
                           const float* __restrict__ sum, const float* __restrict__ sq,
                           const float* __restrict__ gamma, const float* __restrict__ beta,
                           float* __restrict__ out, int total) {
  const int idx = blockIdx.x * blockDim.x + threadIdx.x;
  if (idx >= total) return;
  const int c = idx & 63;
  const float invN = 1.0f / (float)N_NODES;
  const float mu  = sum[c] * invN;
  const float var = fmaxf(sq[c] * invN - mu * mu, 0.0f);   // biased variance
  const float y = (in[idx] - mu) * rsqrtf(var + BN_EPS) * gamma[c] + beta[c];
  out[idx] = fmaxf(y, 0.0f);
}

// ---------------- FC + sigmoid ----------------
__global__ void k_fc(const float* __restrict__ h, const float* __restrict__ Wfc,
                     const float* __restrict__ bfc, float* __restrict__ out, int n) {
  const int i = blockIdx.x * blockDim.x + threadIdx.x;
  if (i >= n) return;
  float acc = bfc[0];
#pragma unroll
  for (int k = 0; k < HD; ++k) acc = fmaf(h[i * HD + k], Wfc[k], acc);
  out[i] = 1.0f / (1.0f + __expf(-acc));
}

extern "C" void kernel_launch(void* const* d_in, const int* in_sizes, int n_in,
                              void* d_out, int out_size, void* d_ws, size_t ws_size,
                              hipStream_t stream) {
  const float* x         = (const float*)d_in[0];
  const float* edge_attr = (const float*)d_in[1];
  const float* We1       = (const float*)d_in[2];
  const float* be1       = (const float*)d_in[3];
  const float* root1     = (const float*)d_in[4];
  const float* b1        = (const float*)d_in[5];
  const float* g1        = (const float*)d_in[6];
  const float* beta1     = (const float*)d_in[7];
  const float* We2       = (const float*)d_in[8];
  const float* be2       = (const float*)d_in[9];
  const float* root2     = (const float*)d_in[10];
  const float* b2        = (const float*)d_in[11];
  const float* g2        = (const float*)d_in[12];
  const float* beta2     = (const float*)d_in[13];
  const float* Wfc       = (const float*)d_in[14];
  const float* bfc       = (const float*)d_in[15];
  const int*   eidx      = (const int*)d_in[16];
  float* out = (float*)d_out;

  // workspace layout (floats):
  //   buf0   [N*64]  h1_pre, then h2_pre
  //   buf1   [N*64]  h1, then h2
  //   stats  [256]   sum1|sq1|sum2|sq2
  //   pack1  [8*1408]  packed We1+be1
  //   pack2  [8*4096]  packed We2+be2
  float* buf0  = (float*)d_ws;
  float* buf1  = buf0 + (size_t)N_NODES * HD;
  float* stats = buf1 + (size_t)N_NODES * HD;
  float* pack1 = stats + 256;
  float* pack2 = pack1 + 8 * (FN * HD);

  const int rowTiles   = N_NODES / 16;               // 3125 (exact)
  const int gemmBlocks = (rowTiles + 7) / 8;         // 8 waves/block
  const int edgeTiles  = N_EDGES / 16;               // 6250 (exact)
  const int edgeBlocks = (edgeTiles + 7) / 8;
  const int elemBlocks = (N_NODES * HD + 255) / 256;

  k_zero<<<1, 256, 0, stream>>>(stats, 256);
  k_pack_we<<<(8 * FN * HD + 255) / 256, 256, 0, stream>>>(We1, be1, pack1, FN * HD);
  k_pack_we<<<(8 * HD * HD + 255) / 256, 256, 0, stream>>>(We2, be2, pack2, HD * HD);

  // ---- layer 1: h1 = relu(BN(segsum(msg) + x@root1 + b1)) ----
  k_root_gemm<FN><<<gemmBlocks, 256, 0, stream>>>(x, root1, b1, buf0, N_NODES);
  k_edge_msg<FN><<<edgeBlocks, 256, 0, stream>>>(x, edge_attr, eidx, pack1, buf0, N_EDGES);
  k_bn_stats<<<128, 256, 0, stream>>>(buf0, stats, stats + 64, N_NODES);
  k_bn_apply<<<elemBlocks, 256, 0, stream>>>(buf0, stats, stats + 64, g1, beta1, buf1, N_NODES * HD);

  // ---- layer 2 (reuse buf0 as h2_pre) ----
  k_root_gemm<HD><<<gemmBlocks, 256, 0, stream>>>(buf1, root2, b2, buf0, N_NODES);
  k_edge_msg<HD><<<edgeBlocks, 256, 0, stream>>>(buf1, edge_attr, eidx, pack2, buf0, N_EDGES);
  k_bn_stats<<<128, 256, 0, stream>>>(buf0, stats + 128, stats + 192, N_NODES);
  k_bn_apply<<<elemBlocks, 256, 0, stream>>>(buf0, stats + 128, stats + 192, g2, beta2, buf1, N_NODES * HD);

  // ---- readout ----
  k_fc<<<(N_NODES + 255) / 256, 256, 0, stream>>>(buf1, Wfc, bfc, out, N_NODES);
}